// DrugGAT_55800215110123
// MI455X (gfx1250) — compile-verified
//
#include <hip/hip_runtime.h>
#include <math.h>

// ---------------- problem constants (from reference) ----------------
#define N_NODES   50000
#define N_EDGES   400000
#define IN_FEATS  128
#define HEADS     4
#define DIM       64
#define HD        (HEADS * DIM)     // 256
#define N_GRAPHS  2000
#define NEG_SLOPE 0.2f

typedef float v2f __attribute__((ext_vector_type(2)));
typedef float v8f __attribute__((ext_vector_type(8)));

// d_out layout (floats): x1 [50000*64] | h [2000*64] | score [400000*4]
#define OUT_X1_OFF    0
#define OUT_H_OFF     (N_NODES * DIM)                    // 3,200,000
#define OUT_SCORE_OFF (OUT_H_OFF + N_GRAPHS * DIM)       // 3,328,000

// workspace layout (floats)
#define WS_FS_OFF     0                                   // 50000*256
#define WS_FD_OFF     (WS_FS_OFF + N_NODES * HD)          // 12,800,000
#define WS_E_OFF      (WS_FD_OFF + N_NODES * HD)          // 25,600,000
#define WS_M_OFF      (WS_E_OFF + N_EDGES * HEADS)        // 27,200,000
#define WS_DEN_OFF    (WS_M_OFF + N_NODES * HEADS)        // 27,400,000
#define WS_CNT_OFF    (WS_DEN_OFF + N_NODES * HEADS)      // 27,600,000

// ---------------- K0: init ----------------
__global__ void __launch_bounds__(256) gat_init(float* __restrict__ x1,
                                                float* __restrict__ hsum,
                                                float* __restrict__ m,
                                                float* __restrict__ denom,
                                                float* __restrict__ cnt) {
    int i = blockIdx.x * blockDim.x + threadIdx.x;
    if (i < N_NODES * DIM)   x1[i] = 0.0f;
    if (i < N_GRAPHS * DIM)  hsum[i] = 0.0f;
    if (i < N_NODES * HEADS) { m[i] = -INFINITY; denom[i] = 0.0f; }
    if (i < N_GRAPHS)        cnt[i] = 0.0f;
}

// ---------------- K1: dual GEMM via V_WMMA_F32_16X16X4_F32 ----------------
// fs = atom @ W_src + b_src ; fd = atom @ W_dst + b_dst
// Block = 512 threads = 16 waves. blockIdx.x = M-tile (3125 tiles of 16 rows).
// Each wave owns one 16-column N-tile and computes BOTH outputs, reusing the
// A fragment (atom tile staged once in LDS per block).
__global__ void __launch_bounds__(512) gat_gemm(const float* __restrict__ atom,
                                                const float* __restrict__ Ws,
                                                const float* __restrict__ bs,
                                                const float* __restrict__ Wd,
                                                const float* __restrict__ bd,
                                                float* __restrict__ fs,
                                                float* __restrict__ fd) {
    __shared__ float As[16][IN_FEATS + 4];   // +4 pad: conflict-free column reads
    const int m0  = blockIdx.x * 16;
    const int tid = threadIdx.x;

    // cooperative, coalesced load of the 16x128 atom tile
    for (int i = tid; i < 16 * IN_FEATS; i += 512) {
        int r = i >> 7, c = i & (IN_FEATS - 1);
        As[r][c] = atom[(m0 + r) * IN_FEATS + c];
    }
    __syncthreads();

    const int wave = tid >> 5;        // 0..15 -> N tile
    const int lane = tid & 31;
    const int half = lane >> 4;       // lanes 16..31 carry K+2,K+3 (A) / rows K+2,K+3 (B)
    const int l16  = lane & 15;
    const int n0   = wave * 16;

    v8f accS = {};
    v8f accD = {};

    for (int k = 0; k < IN_FEATS; k += 4) {
        // A 16x4 fp32 fragment: lane m=l16; VGPR0=K(k+2*half), VGPR1=K(k+2*half+1)
        v2f a;
        a.x = As[l16][k + 2 * half + 0];
        a.y = As[l16][k + 2 * half + 1];
        // B 4x16 fragment: VGPR v holds row K=(k + v + 2*half), cols n0+l16
        v2f bS, bD;
        const int r0 = (k + 2 * half + 0) * HD + n0 + l16;
        const int r1 = (k + 2 * half + 1) * HD + n0 + l16;
        bS.x = Ws[r0]; bS.y = Ws[r1];
        bD.x = Wd[r0]; bD.y = Wd[r1];

        accS = __builtin_amdgcn_wmma_f32_16x16x4_f32(
            false, a, false, bS, (short)0, accS, false, false);
        accD = __builtin_amdgcn_wmma_f32_16x16x4_f32(
            false, a, false, bD, (short)0, accD, false, false);
    }

    // C/D layout: VGPR v -> row m0 + v + 8*half, col n0 + l16
    const int col = n0 + l16;
    const float biasS = bs[col];
    const float biasD = bd[col];
#pragma unroll
    for (int v = 0; v < 8; ++v) {
        const int row = m0 + v + 8 * half;
        fs[row * HD + col] = accS[v] + biasS;
        fd[row * HD + col] = accD[v] + biasD;
    }
}

// ---------------- float atomic max via sign-split trick ----------------
__device__ __forceinline__ void atomicMaxFloat(float* addr, float val) {
    if (val >= 0.0f)
        atomicMax((int*)addr, __float_as_int(val));
    else
        atomicMin((unsigned int*)addr, __float_as_uint(val));
}

// ---------------- K2: e[edge,h] = sum_d lrelu(fs[src]+fd[dst])*attn ; seg-max ----
// wave per edge; 8 coalesced lane-loads cover the 256-wide feature row.
__global__ void __launch_bounds__(256) gat_edge_e(const float* __restrict__ fs,
                                                  const float* __restrict__ fd,
                                                  const float* __restrict__ attn,
                                                  const int* __restrict__ src,
                                                  const int* __restrict__ dst,
                                                  float* __restrict__ e_buf,
                                                  float* __restrict__ m_buf) {
    const int edge = (blockIdx.x * blockDim.x + threadIdx.x) >> 5;
    const int lane = threadIdx.x & 31;
    if (edge >= N_EDGES) return;
    const int s = src[edge];
    const int d = dst[edge];

    float acc[HEADS] = {0.f, 0.f, 0.f, 0.f};
#pragma unroll
    for (int i = 0; i < 8; ++i) {
        const int j = lane + 32 * i;             // head = i>>1
        float v = fs[s * HD + j] + fd[d * HD + j];
        v = (v > 0.0f) ? v : v * NEG_SLOPE;
        acc[i >> 1] += v * attn[j];
    }
#pragma unroll
    for (int h = 0; h < HEADS; ++h) {
        float r = acc[h];
#pragma unroll
        for (int off = 16; off > 0; off >>= 1) r += __shfl_xor(r, off, 32);
        if (lane == 0) {
            e_buf[edge * HEADS + h] = r;
            atomicMaxFloat(&m_buf[d * HEADS + h], r);
        }
    }
}

// ---------------- K3: ex = exp(e - m[dst]); denom += ex; stash ex in score ----
__global__ void __launch_bounds__(256) gat_exp(const float* __restrict__ e_buf,
                                               const float* __restrict__ m_buf,
                                               const int* __restrict__ dst,
                                               float* __restrict__ denom,
                                               float* __restrict__ score) {
    const int i = blockIdx.x * blockDim.x + threadIdx.x;
    if (i >= N_EDGES * HEADS) return;
    const int edge = i >> 2;
    const int h    = i & 3;
    const int d    = dst[edge];
    const float ex = expf(e_buf[i] - m_buf[d * HEADS + h]);
    score[i] = ex;
    atomicAdd(&denom[d * HEADS + h], ex);
}

// ---------------- K4: alpha + head-averaged message scatter into x1 ----------
// wave per edge.
__global__ void __launch_bounds__(256) gat_scatter(const float* __restrict__ fs,
                                                   const int* __restrict__ src,
                                                   const int* __restrict__ dst,
                                                   const float* __restrict__ denom,
                                                   float* __restrict__ score,
                                                   float* __restrict__ x1) {
    const int edge = (blockIdx.x * blockDim.x + threadIdx.x) >> 5;
    const int lane = threadIdx.x & 31;
    if (edge >= N_EDGES) return;
    const int s = src[edge];
    const int d = dst[edge];

    float alpha[HEADS];
#pragma unroll
    for (int h = 0; h < HEADS; ++h)
        alpha[h] = score[edge * HEADS + h] / denom[d * HEADS + h];
    if (lane < HEADS) score[edge * HEADS + lane] = alpha[lane];

#pragma unroll
    for (int half = 0; half < 2; ++half) {
        const int dpos = lane + 32 * half;       // 0..63
        float v = 0.0f;
#pragma unroll
        for (int h = 0; h < HEADS; ++h)
            v += fs[s * HD + h * DIM + dpos] * alpha[h];
        atomicAdd(&x1[d * DIM + dpos], 0.25f * v);
    }
}

// ---------------- K5: per-graph sums + counts (wave per node) ----------------
__global__ void __launch_bounds__(256) gat_pool(const float* __restrict__ x1,
                                                const int* __restrict__ graph_ids,
                                                float* __restrict__ hsum,
                                                float* __restrict__ cnt) {
    const int node = (blockIdx.x * blockDim.x + threadIdx.x) >> 5;
    const int lane = threadIdx.x & 31;
    if (node >= N_NODES) return;
    const int g = graph_ids[node];
#pragma unroll
    for (int half = 0; half < 2; ++half) {
        const int dpos = lane + 32 * half;
        atomicAdd(&hsum[g * DIM + dpos], x1[node * DIM + dpos]);
    }
    if (lane == 0) atomicAdd(&cnt[g], 1.0f);
}

// ---------------- K6: h /= max(cnt, 1) ----------------
__global__ void __launch_bounds__(256) gat_norm(float* __restrict__ hsum,
                                                const float* __restrict__ cnt) {
    const int i = blockIdx.x * blockDim.x + threadIdx.x;
    if (i >= N_GRAPHS * DIM) return;
    float c = cnt[i >> 6];
    c = (c > 1.0f) ? c : 1.0f;
    hsum[i] /= c;
}

// ---------------- launcher ----------------
extern "C" void kernel_launch(void* const* d_in, const int* in_sizes, int n_in,
                              void* d_out, int out_size, void* d_ws, size_t ws_size,
                              hipStream_t stream) {
    const float* atom      = (const float*)d_in[0];
    const float* W_src     = (const float*)d_in[1];
    const float* b_src     = (const float*)d_in[2];
    const float* W_dst     = (const float*)d_in[3];
    const float* b_dst     = (const float*)d_in[4];
    const float* attn      = (const float*)d_in[5];   // HEADS*DIM flat = 256
    const int*   src       = (const int*)d_in[6];
    const int*   dst       = (const int*)d_in[7];
    const int*   graph_ids = (const int*)d_in[8];

    float* out   = (float*)d_out;
    float* x1    = out + OUT_X1_OFF;
    float* h     = out + OUT_H_OFF;
    float* score = out + OUT_SCORE_OFF;

    float* ws    = (float*)d_ws;
    float* fs    = ws + WS_FS_OFF;
    float* fd    = ws + WS_FD_OFF;
    float* e_buf = ws + WS_E_OFF;
    float* m_buf = ws + WS_M_OFF;
    float* denom = ws + WS_DEN_OFF;
    float* cnt   = ws + WS_CNT_OFF;

    // K0: init (covers largest range: x1, 3.2M elements)
    gat_init<<<(N_NODES * DIM + 255) / 256, 256, 0, stream>>>(x1, h, m_buf, denom, cnt);

    // K1: WMMA GEMMs. 3125 M-tiles x (16 waves = all 16 N-tiles) per block.
    gat_gemm<<<N_NODES / 16, 512, 0, stream>>>(atom, W_src, b_src, W_dst, b_dst, fs, fd);

    // K2: edge logits + segment max. 8 edges per 256-thread block.
    gat_edge_e<<<N_EDGES / 8, 256, 0, stream>>>(fs, fd, attn, src, dst, e_buf, m_buf);

    // K3: exp + denom
    gat_exp<<<(N_EDGES * HEADS + 255) / 256, 256, 0, stream>>>(e_buf, m_buf, dst, denom, score);

    // K4: alpha + message scatter
    gat_scatter<<<N_EDGES / 8, 256, 0, stream>>>(fs, src, dst, denom, score, x1);

    // K5: graph pooling sums
    gat_pool<<<(N_NODES + 7) / 8, 256, 0, stream>>>(x1, graph_ids, h, cnt);

    // K6: normalize graph means
    gat_norm<<<(N_GRAPHS * DIM + 255) / 256, 256, 0, stream>>>(h, cnt);
}